// GAT_47107201302624
// MI455X (gfx1250) — compile-verified
//
#include <hip/hip_runtime.h>
#include <hip/hip_bf16.h>
#include <math.h>

// ---- problem constants (from reference) ----
#define N_NODES 100000
#define IN_CH   512
#define HEADS1  8
#define CH1     8
#define H1C     64          // HEADS1*CH1
#define OUT_CH  16
#define E_RAW   3200000
#define E_TOT   (E_RAW + N_NODES)   // edges + self loops
#define NEG_SLOPE 0.2f

#define M_TILES (N_NODES / 16)      // 6250 exactly
#define KCHUNK  128                 // K-tile staged in LDS (128x64 f32 = 32 KB)

typedef __attribute__((ext_vector_type(2))) float v2f;
typedef __attribute__((ext_vector_type(8))) float v8f;

// ---- order-preserving float<->uint transform for atomicMax on floats ----
__device__ __forceinline__ unsigned enc_f32(float f) {
  unsigned u = __float_as_uint(f);
  return (u & 0x80000000u) ? ~u : (u | 0x80000000u);
}
__device__ __forceinline__ float dec_f32(unsigned u) {
  return __uint_as_float((u & 0x80000000u) ? (u & 0x7fffffffu) : ~u);
}
#define ENC_NEG_INF 0x007FFFFFu     // enc_f32(-inf)

// ---------------- init: zero accumulators, -inf the maxes ----------------
__global__ void k_init(unsigned* __restrict__ m1, float* __restrict__ den1,
                       float* __restrict__ acc1, unsigned* __restrict__ m2,
                       float* __restrict__ den2, float* __restrict__ acc2) {
  int i = blockIdx.x * 256 + threadIdx.x;
  if (i < N_NODES * HEADS1) { m1[i] = ENC_NEG_INF; den1[i] = 0.0f; }
  if (i < N_NODES)          { m2[i] = ENC_NEG_INF; den2[i] = 0.0f; }
  if (i < N_NODES * H1C)    acc1[i] = 0.0f;
  if (i < N_NODES * OUT_CH) acc2[i] = 0.0f;
}

// ---------------- GEMM1: h1[N,64] = x[N,512] @ W1[512,64] ----------------
// One wave32 computes a 16(M) x 64(N) stripe with V_WMMA_F32_16X16X4_F32.
// A frag (16x4 f32): lane<16 -> row=lane, {K+0,K+1}; lane>=16 -> row=lane-16, {K+2,K+3}
// B frag (4x16 f32): mirrored: col=lane&15, rows {kb, kb+1}, kb=(lane>>4)*2.
// W1 is staged through LDS in 32KB K-chunks, stored as row-pair float2 so a
// B fragment is a single ds_load_b64.
__global__ __launch_bounds__(128) void k_gemm1(const float* __restrict__ x,
                                               const float* __restrict__ W1,
                                               float* __restrict__ h1) {
  __shared__ v2f sW[(KCHUNK / 2) * H1C];      // 64*64 v2f = 32 KB

  const int lane  = threadIdx.x & 31;
  const int wave  = threadIdx.x >> 5;
  const int mtile = blockIdx.x * 4 + wave;
  const bool valid = (mtile < M_TILES);
  const int mt  = valid ? mtile : (M_TILES - 1);   // keep barriers uniform
  const int l15 = lane & 15;
  const int kb  = (lane >> 4) * 2;            // 0 or 2
  const int hl  = kb >> 1;                    // 0 or 1 (row-pair select)
  const float* xrow = x + (size_t)(mt * 16 + l15) * IN_CH;

  v8f c0 = {}, c1 = {}, c2 = {}, c3 = {};
  for (int k0 = 0; k0 < IN_CH; k0 += KCHUNK) {
    // stage W1[k0 .. k0+127][0..63] as {row 2p, row 2p+1} pairs
    for (int j = threadIdx.x; j < (KCHUNK / 2) * H1C; j += 128) {
      const int p = j >> 6, col = j & 63;
      const float* w = W1 + (size_t)(k0 + 2 * p) * H1C + col;
      sW[j] = v2f{ w[0], w[H1C] };
    }
    __syncthreads();
#pragma unroll 4
    for (int kk = 0; kk < KCHUNK; kk += 4) {
      v2f a = *(const v2f*)(xrow + k0 + kk + kb);     // 8B aligned (kb even)
      const v2f* bp = sW + ((kk >> 1) + hl) * H1C + l15;
      v2f b0 = bp[0], b1 = bp[16], b2 = bp[32], b3 = bp[48];   // ds_load_b64 x4
      c0 = __builtin_amdgcn_wmma_f32_16x16x4_f32(false, a, false, b0, (short)0, c0, false, false);
      c1 = __builtin_amdgcn_wmma_f32_16x16x4_f32(false, a, false, b1, (short)0, c1, false, false);
      c2 = __builtin_amdgcn_wmma_f32_16x16x4_f32(false, a, false, b2, (short)0, c2, false, false);
      c3 = __builtin_amdgcn_wmma_f32_16x16x4_f32(false, a, false, b3, (short)0, c3, false, false);
    }
    __syncthreads();
  }
  if (!valid) return;
  // C/D layout: VGPR r, lanes 0-15 -> M=r, lanes 16-31 -> M=r+8; N=lane&15
  float* out = h1 + (size_t)mt * 16 * H1C;
#pragma unroll
  for (int r = 0; r < 8; ++r) {
    const int m = r + (lane >> 4) * 8;
    out[(size_t)m * H1C +  0 + l15] = c0[r];
    out[(size_t)m * H1C + 16 + l15] = c1[r];
    out[(size_t)m * H1C + 32 + l15] = c2[r];
    out[(size_t)m * H1C + 48 + l15] = c3[r];
  }
}

// ---------------- per (node,head): e_src/e_dst dot products ----------------
__global__ void k_att1(const float* __restrict__ h1, const float* __restrict__ a_src,
                       const float* __restrict__ a_dst, float* __restrict__ es,
                       float* __restrict__ ed) {
  int i = blockIdx.x * 256 + threadIdx.x;     // i = node*8 + head
  if (i >= N_NODES * HEADS1) return;
  const int h = i & 7;
  const float* hv = h1 + (size_t)i * CH1;     // h1[n][h][c] contiguous
  float s = 0.0f, d = 0.0f;
#pragma unroll
  for (int c = 0; c < CH1; ++c) {
    const float v = hv[c];
    s += v * a_src[h * CH1 + c];
    d += v * a_dst[h * CH1 + c];
  }
  es[i] = s; ed[i] = d;
}

__device__ __forceinline__ void edge_sd(const int* __restrict__ ei, int e, int& s, int& d) {
  if (e < E_RAW) { s = ei[e]; d = ei[E_RAW + e]; }
  else           { s = d = e - E_RAW; }
}

// ---------------- layer1 edge pass A: segment max of logits ----------------
__global__ void k_edge1_max(const int* __restrict__ ei, const float* __restrict__ es,
                            const float* __restrict__ ed, unsigned* __restrict__ m) {
  int t = blockIdx.x * 256 + threadIdx.x;     // t = edge*8 + head (26.4M < 2^31)
  if (t >= E_TOT * HEADS1) return;
  const int e = t >> 3, h = t & 7;
  int s, d; edge_sd(ei, e, s, d);
  float lg = es[s * 8 + h] + ed[d * 8 + h];
  lg = lg > 0.0f ? lg : NEG_SLOPE * lg;
  atomicMax(&m[d * 8 + h], enc_f32(lg));
}

// ------- layer1 edge pass B: w=exp(logit-m); accumulate denom + w*h1[src] -------
__global__ void k_edge1_sum(const int* __restrict__ ei, const float* __restrict__ es,
                            const float* __restrict__ ed, const unsigned* __restrict__ m,
                            const float* __restrict__ h1, float* __restrict__ den,
                            float* __restrict__ acc) {
  int t = blockIdx.x * 256 + threadIdx.x;
  if (t >= E_TOT * HEADS1) return;
  const int e = t >> 3, h = t & 7;
  int s, d; edge_sd(ei, e, s, d);
  float lg = es[s * 8 + h] + ed[d * 8 + h];
  lg = lg > 0.0f ? lg : NEG_SLOPE * lg;
  float mv = dec_f32(m[d * 8 + h]);
  if (isinf(mv)) mv = 0.0f;                   // matches reference where(isfinite)
  const float w = expf(lg - mv);
  atomicAdd(&den[d * 8 + h], w);
  const float* hs = h1 + (size_t)s * H1C + h * CH1;
  float* ad = acc + (size_t)d * H1C + h * CH1;
#pragma unroll
  for (int c = 0; c < CH1; ++c) atomicAdd(&ad[c], w * hs[c]);
}

// ------- normalize by denom, add bias, ELU (in place: acc1 -> h2) -------
__global__ void k_norm1(float* __restrict__ acc, const float* __restrict__ den,
                        const float* __restrict__ b1) {
  int i = blockIdx.x * 256 + threadIdx.x;
  if (i >= N_NODES * H1C) return;
  const int n = i >> 6, hc = i & 63, h = hc >> 3;
  float v = acc[i] / den[n * HEADS1 + h] + b1[hc];
  acc[i] = v > 0.0f ? v : expm1f(v);          // ELU
}

// ---------------- GEMM2: h3[N,16] = h2[N,64] @ W2[64,16] ----------------
__global__ __launch_bounds__(128) void k_gemm2(const float* __restrict__ h2,
                                               const float* __restrict__ W2,
                                               float* __restrict__ h3) {
  __shared__ v2f sW[(H1C / 2) * OUT_CH];      // 512 v2f = 4 KB

  const int lane  = threadIdx.x & 31;
  const int wave  = threadIdx.x >> 5;
  const int mtile = blockIdx.x * 4 + wave;
  const bool valid = (mtile < M_TILES);
  const int mt  = valid ? mtile : (M_TILES - 1);
  const int l15 = lane & 15;
  const int kb  = (lane >> 4) * 2;
  const int hl  = kb >> 1;

  for (int j = threadIdx.x; j < (H1C / 2) * OUT_CH; j += 128) {
    const int p = j >> 4, col = j & 15;
    const float* w = W2 + (size_t)(2 * p) * OUT_CH + col;
    sW[j] = v2f{ w[0], w[OUT_CH] };
  }
  __syncthreads();

  const float* hrow = h2 + (size_t)(mt * 16 + l15) * H1C;
  v8f c = {};
#pragma unroll
  for (int k = 0; k < H1C; k += 4) {
    v2f a = *(const v2f*)(hrow + k + kb);
    v2f b = sW[((k >> 1) + hl) * OUT_CH + l15];
    c = __builtin_amdgcn_wmma_f32_16x16x4_f32(false, a, false, b, (short)0, c, false, false);
  }
  if (!valid) return;
  float* out = h3 + (size_t)mt * 16 * OUT_CH;
#pragma unroll
  for (int r = 0; r < 8; ++r) {
    const int mm = r + (lane >> 4) * 8;
    out[(size_t)mm * OUT_CH + l15] = c[r];
  }
}

// ---------------- per node: layer-2 attention dot products ----------------
__global__ void k_att2(const float* __restrict__ h3, const float* __restrict__ a_src,
                       const float* __restrict__ a_dst, float* __restrict__ es,
                       float* __restrict__ ed) {
  int n = blockIdx.x * 256 + threadIdx.x;
  if (n >= N_NODES) return;
  const float* hv = h3 + (size_t)n * OUT_CH;
  float s = 0.0f, d = 0.0f;
#pragma unroll
  for (int c = 0; c < OUT_CH; ++c) { s += hv[c] * a_src[c]; d += hv[c] * a_dst[c]; }
  es[n] = s; ed[n] = d;
}

__global__ void k_edge2_max(const int* __restrict__ ei, const float* __restrict__ es,
                            const float* __restrict__ ed, unsigned* __restrict__ m) {
  int e = blockIdx.x * 256 + threadIdx.x;
  if (e >= E_TOT) return;
  int s, d; edge_sd(ei, e, s, d);
  float lg = es[s] + ed[d];
  lg = lg > 0.0f ? lg : NEG_SLOPE * lg;
  atomicMax(&m[d], enc_f32(lg));
}

__global__ void k_edge2_sum(const int* __restrict__ ei, const float* __restrict__ es,
                            const float* __restrict__ ed, const unsigned* __restrict__ m,
                            const float* __restrict__ h3, float* __restrict__ den,
                            float* __restrict__ acc) {
  int e = blockIdx.x * 256 + threadIdx.x;
  if (e >= E_TOT) return;
  int s, d; edge_sd(ei, e, s, d);
  float lg = es[s] + ed[d];
  lg = lg > 0.0f ? lg : NEG_SLOPE * lg;
  float mv = dec_f32(m[d]);
  if (isinf(mv)) mv = 0.0f;
  const float w = expf(lg - mv);
  atomicAdd(&den[d], w);
  const float* hs = h3 + (size_t)s * OUT_CH;
  float* ad = acc + (size_t)d * OUT_CH;
#pragma unroll
  for (int c = 0; c < OUT_CH; ++c) atomicAdd(&ad[c], w * hs[c]);
}

// ------- final: normalize, +bias, log_softmax over 16 channels -------
__global__ void k_out(const float* __restrict__ acc, const float* __restrict__ den,
                      const float* __restrict__ b2, float* __restrict__ out) {
  int n = blockIdx.x * 256 + threadIdx.x;
  if (n >= N_NODES) return;
  const float inv = 1.0f / den[n];
  float v[OUT_CH], mx = -3.4e38f;
#pragma unroll
  for (int c = 0; c < OUT_CH; ++c) { v[c] = acc[(size_t)n * OUT_CH + c] * inv + b2[c]; mx = fmaxf(mx, v[c]); }
  float ss = 0.0f;
#pragma unroll
  for (int c = 0; c < OUT_CH; ++c) ss += expf(v[c] - mx);
  const float lse = mx + logf(ss);
#pragma unroll
  for (int c = 0; c < OUT_CH; ++c) out[(size_t)n * OUT_CH + c] = v[c] - lse;
}

extern "C" void kernel_launch(void* const* d_in, const int* in_sizes, int n_in,
                              void* d_out, int out_size, void* d_ws, size_t ws_size,
                              hipStream_t stream) {
  const float* x   = (const float*)d_in[0];
  const int*   ei  = (const int*)  d_in[1];
  const float* W1  = (const float*)d_in[2];
  const float* a1s = (const float*)d_in[3];
  const float* a1d = (const float*)d_in[4];
  const float* b1  = (const float*)d_in[5];
  const float* W2  = (const float*)d_in[6];
  const float* a2s = (const float*)d_in[7];
  const float* a2d = (const float*)d_in[8];
  const float* b2  = (const float*)d_in[9];
  float* out = (float*)d_out;

  // workspace carve-up (floats), ~78 MB total
  char* p = (char*)d_ws;
  auto take = [&](size_t elems) { void* r = p; p += ((elems * 4 + 255) & ~(size_t)255); return r; };
  float*    h1   = (float*)   take((size_t)N_NODES * H1C);
  float*    es1  = (float*)   take((size_t)N_NODES * HEADS1);
  float*    ed1  = (float*)   take((size_t)N_NODES * HEADS1);
  unsigned* m1   = (unsigned*)take((size_t)N_NODES * HEADS1);
  float*    den1 = (float*)   take((size_t)N_NODES * HEADS1);
  float*    acc1 = (float*)   take((size_t)N_NODES * H1C);     // becomes h2 after norm1
  float*    h3   = (float*)   take((size_t)N_NODES * OUT_CH);
  float*    es2  = (float*)   take((size_t)N_NODES);
  float*    ed2  = (float*)   take((size_t)N_NODES);
  unsigned* m2   = (unsigned*)take((size_t)N_NODES);
  float*    den2 = (float*)   take((size_t)N_NODES);
  float*    acc2 = (float*)   take((size_t)N_NODES * OUT_CH);

  const int gInit  = (N_NODES * H1C + 255) / 256;          // covers largest buffer
  const int gGemm  = (M_TILES + 3) / 4;                    // 4 waves (tiles) per block
  const int gNH    = (N_NODES * HEADS1 + 255) / 256;
  const int gEH    = (E_TOT * HEADS1 + 255) / 256;
  const int gE     = (E_TOT + 255) / 256;
  const int gN     = (N_NODES + 255) / 256;

  k_init     <<<gInit, 256, 0, stream>>>(m1, den1, acc1, m2, den2, acc2);
  k_gemm1    <<<gGemm, 128, 0, stream>>>(x, W1, h1);
  k_att1     <<<gNH,   256, 0, stream>>>(h1, a1s, a1d, es1, ed1);
  k_edge1_max<<<gEH,   256, 0, stream>>>(ei, es1, ed1, m1);
  k_edge1_sum<<<gEH,   256, 0, stream>>>(ei, es1, ed1, m1, h1, den1, acc1);
  k_norm1    <<<gInit, 256, 0, stream>>>(acc1, den1, b1);  // acc1 -> h2 in place
  k_gemm2    <<<gGemm, 128, 0, stream>>>(acc1, W2, h3);
  k_att2     <<<gN,    256, 0, stream>>>(h3, a2s, a2d, es2, ed2);
  k_edge2_max<<<gE,    256, 0, stream>>>(ei, es2, ed2, m2);
  k_edge2_sum<<<gE,    256, 0, stream>>>(ei, es2, ed2, m2, h3, den2, acc2);
  k_out      <<<gN,    256, 0, stream>>>(acc2, den2, b2, out);
}